// EnFCNet_25898652795510
// MI455X (gfx1250) — compile-verified
//
#include <hip/hip_runtime.h>

// ---------------------------------------------------------------------------
// EnFCNet block for MI455X (gfx1250, wave32, WMMA + Tensor Data Mover).
//
// One workgroup per GO-group g (256 groups). Both stages are 32x256 f32 GEMM
// tiles whose BatchNorm statistics cover exactly that tile, so GEMM + BN +
// ReLU fuse into a single workgroup-local kernel using exact-f32
// V_WMMA_F32_16X16X4_F32 (reference math is f32 -> no downcast).
//
// Data movement: the W tile (256 rows x 32 k, row pitch = K) is DMA'd into
// LDS by the TDM (tensor_load_to_lds) with hardware row padding (+4 dwords)
// and hardware OOB zero-fill for the K tail; double-buffered so the DMA of
// chunk c+1 overlaps the 8 WMMAs of chunk c (TENSORcnt + barrier sync).
// ---------------------------------------------------------------------------

typedef float v2f __attribute__((ext_vector_type(2)));
typedef float v8f __attribute__((ext_vector_type(8)));
typedef unsigned int uint32x4 __attribute__((ext_vector_type(4)));
typedef int int32x4 __attribute__((ext_vector_type(4)));
typedef int int32x8 __attribute__((ext_vector_type(8)));

#define BDIM   1024
#define KC     32                 // K-chunk staged in LDS
#define LDSTR  (KC + 4)           // LDS row stride in floats (TDM pad: +4 dwords)

#if __has_builtin(__builtin_amdgcn_tensor_load_to_lds)
#define HAVE_TDM 1
#else
#define HAVE_TDM 0
#endif

// Issue one TDM 2D tile copy: W[0:256, k0:k0+32] (f32, row pitch K elems)
// -> LDS at lds_off, with +4 dwords padding every 32 dwords (row stride 36).
// tensor_dim0 = K - k0 so the hardware zero-fills the K tail (OOB reads = 0).
__device__ __forceinline__ void tdm_load_w_tile(const float* Bm, int K, int k0,
                                                unsigned lds_off) {
#if HAVE_TDM
    unsigned long long ga =
        (unsigned long long)(const void*)Bm + (unsigned long long)k0 * 4ull;
    uint32x4 g0;
    g0.x = 1u;                                   // count=1 (valid), user mode
    g0.y = lds_off;                              // lds_addr (bytes)
    g0.z = (unsigned)(ga & 0xFFFFFFFFull);       // global_addr[31:0]
    g0.w = (unsigned)((ga >> 32) & 0x1FFFFFFull) // global_addr[56:32]
           | (2u << 30);                         // type = 2 ("image")

    const unsigned td0 = (unsigned)(K - k0);     // remaining cols (tail -> 0)
    int32x8 g1;
    g1[0] = (int)((2u << 16)      // data_size = 4B
                | (1u << 20)      // pad_enable
                | (4u << 22)      // pad_interval: 32 dwords
                | (3u << 25));    // pad_amount:   4 dwords
    g1[1] = (int)((td0 & 0xFFFFu) << 16);        // tensor_dim0[15:0]
    g1[2] = (int)((td0 >> 16) | (256u << 16));   // dim0[31:16] | tensor_dim1
    g1[3] = (int)(32u << 16);                    // tile_dim0 = 32
    g1[4] = (int)256u;                           // tile_dim1 = 256, tile_dim2=0
    g1[5] = (int)(unsigned)K;                    // tensor_dim0_stride[31:0]
    g1[6] = 0;                                   // stride[47:32] | dim1_stride
    g1[7] = 0;
    int32x4 gz4 = {0, 0, 0, 0};                  // groups 2/3: NULL (2D tensor)
    int32x8 gz8 = {0, 0, 0, 0, 0, 0, 0, 0};     // extended group (unused)
    __builtin_amdgcn_tensor_load_to_lds(g0, g1, gz4, gz4, gz8, 0);
#endif
}

template <bool HAS_MASK>
__global__ __launch_bounds__(BDIM)
void enfc_group_gemm_bn_relu(
    const float* __restrict__ A,     // stage1: x (32 x N); stage2: hbuf (G x 32 x 256)
    long a_g_stride,                 // 0 (x shared across g) or 32*256
    int  lda,                        // A row length (= K)
    const float* __restrict__ mask,  // (G x N); unused when !HAS_MASK
    const float* __restrict__ Bm,    // W (256 x K) row-major; row n = output col n
    const float* __restrict__ bias,  // (256)
    const float* __restrict__ gamma, // (G)
    const float* __restrict__ beta,  // (G)
    float* __restrict__ out,
    long out_g_stride, long out_m_stride,
    int K)
{
    __shared__ float Ws[2][256 * LDSTR];   // W tile, row-major, TDM-padded rows
    __shared__ float As[2][32 * LDSTR];    // A tile, mask fused, same row stride
    __shared__ float wsum[32], wsq[32];    // per-wave BN partials

    const int g     = blockIdx.x;
    const int tid   = threadIdx.x;
    const int lane  = tid & 31;
    const int wave  = tid >> 5;
    const int mbase = (wave & 1) * 16;     // M tile (batch rows)
    const int nbase = (wave >> 1) * 16;    // N tile (output cols)
    const int lhalf = lane >> 4;           // 0: lanes 0-15, 1: lanes 16-31
    const int l15   = lane & 15;

    const float* Ag = A + (long)g * a_g_stride;
    const float* Mg = mask + (long)g * lda;   // only dereferenced if HAS_MASK

    const int nchunks = (K + KC - 1) / KC;

    // ---- branchless A-tile stage (32 x 32, GO mask fused, clamped tail) ----
    auto stage_a = [&](int buf, int k0) {
        const int m  = tid >> 5;
        const int k  = tid & 31;
        const int gk = k0 + k;
        const int ck = gk < K ? gk : K - 1;          // clamp, no exec branch
        float v = Ag[(long)m * lda + ck];
        if (HAS_MASK) v *= Mg[ck];
        As[buf][m * LDSTR + k] = gk < K ? v : 0.0f;  // select, not branch
    };

#if !HAVE_TDM
    // Fallback W-tile stage if the TDM builtin is unavailable.
    auto stage_w = [&](int buf, int k0) {
        #pragma unroll
        for (int i = 0; i < 8; ++i) {
            const int p  = tid + i * BDIM;   // 0..8191 = 256 rows x 32 k
            const int n  = p >> 5;
            const int k  = p & 31;
            const int gk = k0 + k;
            Ws[buf][n * LDSTR + k] = gk < K ? Bm[(long)n * K + gk] : 0.0f;
        }
    };
#endif

    // ---- prologue: stage chunk 0 ----
    stage_a(0, 0);
#if HAVE_TDM
    if (wave == 0) {
        tdm_load_w_tile(Bm, K, 0, (unsigned)(unsigned long long)(uintptr_t)&Ws[0][0]);
        __builtin_amdgcn_s_wait_tensorcnt(0);
    }
#else
    stage_w(0, 0);
#endif
    __syncthreads();

    v8f acc = {};
    int buf = 0;
    for (int c = 0; c < nchunks; ++c) {
        const int nbuf = buf ^ 1;
        // ---- kick off chunk c+1: TDM DMA overlaps this chunk's WMMAs ----
        if (c + 1 < nchunks) {
            stage_a(nbuf, (c + 1) * KC);
#if HAVE_TDM
            if (wave == 0)
                tdm_load_w_tile(Bm, K, (c + 1) * KC,
                                (unsigned)(unsigned long long)(uintptr_t)&Ws[nbuf][0]);
#else
            stage_w(nbuf, (c + 1) * KC);
#endif
        }

        // ---- 8 x V_WMMA_F32_16X16X4_F32 on chunk c ----
        const float* as = &As[buf][(mbase + l15) * LDSTR];
        const float* ws = &Ws[buf][(nbase + l15) * LDSTR];
        #pragma unroll
        for (int kk = 0; kk < 8; ++kk) {
            // A 16x4 frag: lanes0-15 K=4kk..+1, lanes16-31 K=4kk+2..+3
            v2f a = *(const v2f*)&as[4 * kk + 2 * lhalf];
            // B 4x16 frag: B[k][n] = W[n][k] -> same float2 pattern, row = n
            v2f b = *(const v2f*)&ws[4 * kk + 2 * lhalf];
            acc = __builtin_amdgcn_wmma_f32_16x16x4_f32(
                false, a, false, b, (short)0, acc, false, false);
        }

#if HAVE_TDM
        if (wave == 0) __builtin_amdgcn_s_wait_tensorcnt(0);
#endif
        __syncthreads();     // chunk c+1 (TDM + A stores) now visible to all
        buf = nbuf;
    }

    // ---- bias, then workgroup-local BatchNorm over the full 32x256 tile ----
    const float bias_n = bias[nbase + l15];
    float s = 0.0f, q = 0.0f;
    #pragma unroll
    for (int r = 0; r < 8; ++r) {
        acc[r] += bias_n;
        s += acc[r];
        q += acc[r] * acc[r];
    }
    // deterministic reduction: intra-wave xor tree, then fixed-order sum of
    // the 32 per-wave partials by every thread.
    #pragma unroll
    for (int off = 16; off > 0; off >>= 1) {
        s += __shfl_xor(s, off, 32);
        q += __shfl_xor(q, off, 32);
    }
    if (lane == 0) { wsum[wave] = s; wsq[wave] = q; }
    __syncthreads();
    float ts = 0.0f, tq = 0.0f;
    #pragma unroll
    for (int w = 0; w < 32; ++w) { ts += wsum[w]; tq += wsq[w]; }

    const float inv_cnt = 1.0f / (32.0f * 256.0f);   // stats over (B, C)
    const float mean = ts * inv_cnt;
    const float var  = tq * inv_cnt - mean * mean;   // biased var (jnp.var)
    const float rstd = __frsqrt_rn(var + 1e-5f);
    const float gm   = gamma[g] * rstd;
    const float bt   = beta[g] - mean * gm;

    // ---- fused scale/shift + ReLU, store straight from accumulators ----
    float* og = out + (long)g * out_g_stride;
    #pragma unroll
    for (int r = 0; r < 8; ++r) {
        const int mrow = mbase + lhalf * 8 + r;  // C/D layout: VGPR r -> M=r / r+8
        float y = gm * acc[r] + bt;
        y = y > 0.0f ? y : 0.0f;
        og[(long)mrow * out_m_stride + nbase + l15] = y;
    }
}

extern "C" void kernel_launch(void* const* d_in, const int* in_sizes, int n_in,
                              void* d_out, int out_size, void* d_ws, size_t ws_size,
                              hipStream_t stream) {
    (void)in_sizes; (void)n_in; (void)out_size; (void)ws_size;
    const float* x      = (const float*)d_in[0];   // (32, 10000)
    const float* mask   = (const float*)d_in[1];   // (256, 10000)
    const float* W1     = (const float*)d_in[2];   // (256, 10000)
    const float* b1     = (const float*)d_in[3];   // (256)
    const float* gamma1 = (const float*)d_in[4];   // (256)
    const float* beta1  = (const float*)d_in[5];   // (256)
    const float* W2     = (const float*)d_in[6];   // (256, 256)
    const float* b2     = (const float*)d_in[7];   // (256)
    const float* gamma2 = (const float*)d_in[8];   // (256)
    const float* beta2  = (const float*)d_in[9];   // (256)

    float* hbuf = (float*)d_ws;                    // (G=256, B=32, HID=256) = 8 MB

    const int G = 256, N = 10000, HID = 256;
    dim3 grid(G), block(BDIM);

    // Stage 1: h = relu(BN1((x*mask_g) @ W1^T + b1)); hbuf g-major so stage-2
    // A loads are contiguous per workgroup.
    enfc_group_gemm_bn_relu<true><<<grid, block, 0, stream>>>(
        x, /*a_g_stride=*/0L, /*lda=*/N, mask, W1, b1, gamma1, beta1,
        hbuf, /*out_g_stride=*/(long)32 * HID, /*out_m_stride=*/HID, /*K=*/N);

    // Stage 2: z = relu(BN2(h @ W2^T + b2)), written as (b, g, z).
    enfc_group_gemm_bn_relu<false><<<grid, block, 0, stream>>>(
        hbuf, /*a_g_stride=*/(long)32 * HID, /*lda=*/HID, /*mask=*/nullptr,
        W2, b2, gamma2, beta2,
        (float*)d_out, /*out_g_stride=*/(long)HID, /*out_m_stride=*/(long)G * HID,
        /*K=*/HID);
}